// BaseRNNRegressor_46179488366844
// MI455X (gfx1250) — compile-verified
//
#include <hip/hip_runtime.h>
#include <hip/hip_bf16.h>

// Problem constants (reference: B,T,F,H = 512,512,256,64; fp32 everywhere)
#define B_DIM 512
#define T_DIM 512
#define F_DIM 256
#define H_DIM 64

// LDS row strides (floats), padded for bank-conflict-free WMMA fragment reads:
//  - A-side (rows across lanes, K split per half-wave): stride 68 -> row*68 mod 64 = 4*row
//  - B-side (cols across lanes, K rows): stride 72 -> 2*72 mod 64 = 16 separates half-waves
#define XS_STRIDE 68
#define WS_STRIDE 72

typedef float v2f __attribute__((ext_vector_type(2)));
typedef float v8f __attribute__((ext_vector_type(8)));
typedef unsigned int v4u __attribute__((ext_vector_type(4)));
typedef unsigned int v8u __attribute__((ext_vector_type(8)));

// D = A(16x4 f32) * B(4x16 f32) + C(16x16 f32)  (codegen-confirmed round 1)
__device__ __forceinline__ v8f wmma_f32x4(v2f a, v2f b, v8f c) {
    return __builtin_amdgcn_wmma_f32_16x16x4_f32(
        false, a, false, b, (short)0, c, false, false);
}

// ---------------------------------------------------------------------------
// Tensor Data Mover: 2D tile load Global -> LDS with optional row padding.
// Builds the D# per CDNA5 ISA and issues
// tensor_load_to_lds via inline asm (toolchain-portable; tracked by TENSORcnt).
//   pad_interval code: 5 -> pad after every 64 DWORDs of data
//   pad_amount   code: 3 -> 4 DWORDs pad (stride 68), 7 -> 8 DWORDs (stride 72)
// ---------------------------------------------------------------------------
__device__ __forceinline__ void tdm_load_2d(
    unsigned int lds_addr, const void* gptr,
    unsigned int tile_d0, unsigned int tile_d1, unsigned int stride_elems,
    unsigned int pad_en, unsigned int pad_interval, unsigned int pad_amount) {
    const unsigned long long ga = (unsigned long long)(uintptr_t)gptr;

    v4u g0;
    g0[0] = 1u;                                   // count=1 (valid), user mode
    g0[1] = lds_addr;                             // LDS byte address
    g0[2] = (unsigned int)ga;                     // global_addr[31:0]
    g0[3] = ((unsigned int)(ga >> 32) & 0x01FFFFFFu) | (2u << 30); // [56:32] | type=2

    v8u g1;
    g1[0] = (2u << 16)                            // data_size = 4 bytes
          | (pad_en << 20) | (pad_interval << 22) | (pad_amount << 25);
    const unsigned int td0 = stride_elems;        // tensor_dim0 = row length
    const unsigned int td1 = 1u << 20;            // tensor_dim1: large (in-bounds)
    g1[1] = (td0 & 0xFFFFu) << 16;                // dim0[15:0]   @ bits 63:48
    g1[2] = (td0 >> 16) | ((td1 & 0xFFFFu) << 16);// dim0[31:16], dim1[15:0]
    g1[3] = (td1 >> 16) | (tile_d0 << 16);        // dim1[31:16], tile_dim0
    g1[4] = tile_d1 & 0xFFFFu;                    // tile_dim1 (tile_dim2 = 0)
    g1[5] = stride_elems;                         // tensor_dim0_stride[31:0]
    g1[6] = 0u;                                   // stride[47:32]=0, dim1_stride lo
    g1[7] = 0u;

    v4u g2 = {}; v4u g3 = {};                     // 2D: higher dims unused/zero
    asm volatile("tensor_load_to_lds %0, %1, %2, %3"
                 :: "s"(g0), "s"(g1), "s"(g2), "s"(g3)
                 : "memory");
}

// ---------------------------------------------------------------------------
// Kernel 1: xp[t,b,h] = sum_f x[b,t,f] * Wx[f,h] + bias[h]   (time-major out)
// Block = 128 threads (4 waves); tile 64 M-rows x 64 N-cols; K in 4 chunks of
// 64 staged into LDS by TDM (wave 0 issues, hardware inserts the bank pad).
// ---------------------------------------------------------------------------
__global__ __launch_bounds__(128) void xp_gemm_kernel(
    const float* __restrict__ x, const float* __restrict__ Wx,
    const float* __restrict__ bias, float* __restrict__ xp) {
    __shared__ float xs[64 * XS_STRIDE];   // x tile  (17 KB, stride 68)
    __shared__ float ws[64 * WS_STRIDE];   // Wx tile (18 KB, stride 72)

    const int tid  = threadIdx.x;
    const int lane = tid & 31;
    const int wave = tid >> 5;
    const int l16  = lane & 15;
    const int hi   = lane >> 4;
    const long mbase = (long)blockIdx.x * 64;

    const unsigned int xs_lds = (unsigned int)(uintptr_t)(void*)&xs[0];
    const unsigned int ws_lds = (unsigned int)(uintptr_t)(void*)&ws[0];

    v8f acc[4] = {};

    for (int kc = 0; kc < 4; ++kc) {
        if (tid < 32) {                    // one wave drives the DMAs
            tdm_load_2d(xs_lds, &x[mbase * F_DIM + kc * 64],
                        64, 64, F_DIM, 1, 5, 3);          // pad -> stride 68
            tdm_load_2d(ws_lds, &Wx[(long)(kc * 64) * H_DIM],
                        64, 64, H_DIM, 1, 5, 7);          // pad -> stride 72
            __builtin_amdgcn_s_wait_tensorcnt(0);
        }
        __syncthreads();                   // publish LDS tiles

        for (int ks = 0; ks < 16; ++ks) {
            const int kk   = ks * 4 + hi * 2;
            const int arow = wave * 16 + l16;
            v2f a;
            a.x = xs[arow * XS_STRIDE + kk];
            a.y = xs[arow * XS_STRIDE + kk + 1];
            for (int n = 0; n < 4; ++n) {
                v2f b;
                b.x = ws[kk * WS_STRIDE + n * 16 + l16];
                b.y = ws[(kk + 1) * WS_STRIDE + n * 16 + l16];
                acc[n] = wmma_f32x4(a, b, acc[n]);
            }
        }
        __syncthreads();                   // done reading before next DMA
    }

    for (int n = 0; n < 4; ++n) {
        const float bb = bias[n * 16 + l16];
        for (int v = 0; v < 8; ++v) {
            const int  mloc = v + hi * 8;
            const long m    = mbase + wave * 16 + mloc;
            const int  b    = (int)(m >> 9);
            const int  t    = (int)(m & (T_DIM - 1));
            xp[((long)t * B_DIM + b) * H_DIM + n * 16 + l16] = acc[n][v] + bb;
        }
    }
}

// ---------------------------------------------------------------------------
// Kernel 2: h_t = tanh(xp_t + h_{t-1} @ Wh); out = h_T @ Wd + bd.
// Block = 128 threads (4 waves), 16 batch rows. Wh B-fragments resident in
// VGPRs; h ping-pongs through LDS. xp tile for step t+1 is TDM-prefetched
// while step t computes (tensorcnt<=1 keeps exactly one DMA in flight).
// ---------------------------------------------------------------------------
__global__ __launch_bounds__(128) void rnn_scan_kernel(
    const float* __restrict__ xp, const float* __restrict__ Wh,
    const float* __restrict__ Wd, const float* __restrict__ bd,
    float* __restrict__ out) {
    __shared__ float h[2][16 * XS_STRIDE];    // hidden state ping-pong (8.7 KB)
    __shared__ float xb[2][16 * XS_STRIDE];   // xp tile double buffer (8.7 KB)

    const int tid  = threadIdx.x;
    const int lane = tid & 31;
    const int wave = tid >> 5;
    const int l16  = lane & 15;
    const int hi   = lane >> 4;
    const int bbase = blockIdx.x * 16;

    const unsigned int xb_lds0 = (unsigned int)(uintptr_t)(void*)&xb[0][0];
    const unsigned int xb_lds1 = (unsigned int)(uintptr_t)(void*)&xb[1][0];

    // Wh B-fragments for this wave's N-tile, resident in 32 VGPRs.
    v2f bfrag[16];
    for (int kt = 0; kt < 16; ++kt) {
        const int krow = kt * 4 + hi * 2;
        bfrag[kt].x = Wh[(krow)     * H_DIM + wave * 16 + l16];
        bfrag[kt].y = Wh[(krow + 1) * H_DIM + wave * 16 + l16];
    }

    // h0 = 0; kick off the DMA for the t=0 xp tile.
    for (int i = tid; i < 2 * 16 * XS_STRIDE; i += 128) (&h[0][0])[i] = 0.0f;
    if (tid < 32) {
        tdm_load_2d(xb_lds0, &xp[(long)bbase * H_DIM], 64, 16, H_DIM, 1, 5, 3);
    }

    for (int t = 0; t < T_DIM; ++t) {
        if (tid < 32) {
            if (t + 1 < T_DIM) {           // prefetch next step's tile
                tdm_load_2d((t & 1) ? xb_lds0 : xb_lds1,
                            &xp[((long)(t + 1) * B_DIM + bbase) * H_DIM],
                            64, 16, H_DIM, 1, 5, 3);
                __builtin_amdgcn_s_wait_tensorcnt(1);   // tile t done, t+1 in flight
            } else {
                __builtin_amdgcn_s_wait_tensorcnt(0);
            }
        }
        __syncthreads();                   // xb[t&1] visible; h reads of t-1 done

        const float* hb  = h[t & 1];
        float*       hn  = h[(t + 1) & 1];
        const float* xbt = xb[t & 1];

        v8f c;
        for (int v = 0; v < 8; ++v) {
            const int mloc = v + hi * 8;
            c[v] = xbt[mloc * XS_STRIDE + wave * 16 + l16];
        }

        for (int kt = 0; kt < 16; ++kt) {
            const int kk = kt * 4 + hi * 2;
            v2f a;
            a.x = hb[l16 * XS_STRIDE + kk];
            a.y = hb[l16 * XS_STRIDE + kk + 1];
            c = wmma_f32x4(a, bfrag[kt], c);
        }

        for (int v = 0; v < 8; ++v) {
            const int mloc = v + hi * 8;
            hn[mloc * XS_STRIDE + wave * 16 + l16] = tanhf(c[v]);
        }
        __syncthreads();                   // h_{t+1} published; xb[t&1] free
    }

    // Final dense: h_512 lives in buffer (512 & 1) == 0.
    if (tid < 16) {
        const float* hf = h[0];
        float acc2 = bd[0];
        for (int k = 0; k < H_DIM; ++k) acc2 += hf[tid * XS_STRIDE + k] * Wd[k];
        out[bbase + tid] = acc2;
    }
}

// ---------------------------------------------------------------------------
extern "C" void kernel_launch(void* const* d_in, const int* in_sizes, int n_in,
                              void* d_out, int out_size, void* d_ws, size_t ws_size,
                              hipStream_t stream) {
    const float* x  = (const float*)d_in[0];   // [B,T,F]
    const float* Wx = (const float*)d_in[1];   // [F,H]
    const float* Wh = (const float*)d_in[2];   // [H,H]
    const float* b  = (const float*)d_in[3];   // [H]
    const float* Wd = (const float*)d_in[4];   // [H,1]
    const float* bd = (const float*)d_in[5];   // [1]
    float* out = (float*)d_out;                // [B,1]

    float* xp = (float*)d_ws;                  // [T,B,H] f32 scratch (64 MB)

    xp_gemm_kernel<<<(B_DIM * T_DIM) / 64, 128, 0, stream>>>(x, Wx, b, xp);
    rnn_scan_kernel<<<B_DIM / 16, 128, 0, stream>>>(xp, Wh, Wd, bd, out);
}